// EnergyInGraph_56341380989636
// MI455X (gfx1250) — compile-verified
//
#include <hip/hip_runtime.h>
#include <cstdint>

// ---------------------------------------------------------------------------
// EnergyInGraph for MI455X (gfx1250).
// Strategy: counting-sort rows by graph id, then one workgroup per graph does
// an atomic-free gather-reduce with register accumulation. The per-graph row
// index list is staged into LDS with CDNA5 async global->LDS loads
// (global_load_async_to_lds_b32 / s_wait_asynccnt).
// ---------------------------------------------------------------------------

#define WG 256
#define IDX_CAP 1024   // LDS-staged row indices per graph (avg ~220, max ~300)

typedef uint32_t __attribute__((address_space(3))) lds_u32_t;

__device__ __forceinline__ uint32_t lds_offset_of(uint32_t* generic_lds_ptr) {
    // generic -> LDS addrspace cast, then ptrtoint gives the 32-bit LDS byte
    // offset the async-load instruction expects in its vdst register.
    return (uint32_t)(uintptr_t)(lds_u32_t*)generic_lds_ptr;
}

__device__ __forceinline__ void async_load_b32_to_lds(uint32_t lds_off, const uint32_t* gaddr) {
    asm volatile("global_load_async_to_lds_b32 %0, %1, off"
                 :: "v"(lds_off), "v"(gaddr)
                 : "memory");
}

__device__ __forceinline__ void wait_asynccnt0() {
    asm volatile("s_wait_asynccnt 0" ::: "memory");
}

// --------------------------- small utility kernels -------------------------

__global__ void zero_u32_kernel(uint32_t* p, int n) {
    int i = blockIdx.x * blockDim.x + threadIdx.x;
    if (i < n) p[i] = 0u;
}

__global__ void zero_f32_kernel(float* p, int n) {
    int i = blockIdx.x * blockDim.x + threadIdx.x;
    if (i < n) p[i] = 0.0f;
}

// --------------------------- phase 1: histogram ----------------------------

__device__ __forceinline__ int term_gid(int t, const int* gid2, const int* gid3,
                                        const int* gid4, int N2, int N3, int G) {
    int g = (t < N2) ? gid2[t]
          : (t < N2 + N3) ? gid3[t - N2]
                          : gid4[t - N2 - N3];
    // clamp defensively so a bad id can never corrupt memory
    g = (g < 0) ? 0 : (g >= G ? G - 1 : g);
    return g;
}

__global__ void hist_kernel(const int* gid2, const int* gid3, const int* gid4,
                            int N2, int N3, int N4, int G, uint32_t* count) {
    const int Nt = N2 + N3 + N4;
    for (int t = blockIdx.x * blockDim.x + threadIdx.x; t < Nt;
         t += gridDim.x * blockDim.x) {
        int g = term_gid(t, gid2, gid3, gid4, N2, N3, G);
        atomicAdd(&count[g], 1u);
    }
}

// --------------------------- phase 2: exclusive scan -----------------------
// Single workgroup; G is tiny (4096).

__global__ __launch_bounds__(WG) void scan_kernel(const uint32_t* count,
                                                  uint32_t* startp, uint32_t* cursor, int G) {
    __shared__ uint32_t ssum[WG];
    __shared__ uint32_t spre[WG];
    const int tid = threadIdx.x;
    const int items = (G + WG - 1) / WG;
    const int base = tid * items;

    uint32_t s = 0;
    for (int j = 0; j < items; ++j) {
        int g = base + j;
        if (g < G) s += count[g];
    }
    ssum[tid] = s;
    __syncthreads();
    if (tid == 0) {
        uint32_t run = 0;
        for (int t = 0; t < WG; ++t) { spre[t] = run; run += ssum[t]; }
    }
    __syncthreads();
    uint32_t run = spre[tid];
    for (int j = 0; j < items; ++j) {
        int g = base + j;
        if (g < G) {
            startp[g] = run;
            cursor[g] = run;   // scatter cursors start at the segment base
            run += count[g];
        }
    }
}

// --------------------------- phase 3: scatter ------------------------------

__global__ void scatter_kernel(const int* gid2, const int* gid3, const int* gid4,
                               int N2, int N3, int N4, int G,
                               uint32_t* cursor, uint32_t* ordered) {
    const int Nt = N2 + N3 + N4;
    for (int t = blockIdx.x * blockDim.x + threadIdx.x; t < Nt;
         t += gridDim.x * blockDim.x) {
        int g = term_gid(t, gid2, gid3, gid4, N2, N3, G);
        uint32_t pos = atomicAdd(&cursor[g], 1u);
        ordered[pos] = (uint32_t)t;   // after this kernel, cursor[g] == end[g]
    }
}

// --------------------------- phase 4: gather-reduce ------------------------
// One workgroup per graph. Lane layout: c = tid & 63 (conformation),
// row-lane = tid >> 6 (4 row streams). Each wave handles a single row at a
// time -> x loads are fully coalesced, per-row params are broadcast. Row
// index list is staged to LDS via async global->LDS DMA.

__global__ __launch_bounds__(WG) void gather_reduce_kernel(
        const float* __restrict__ x2, const float* __restrict__ k2, const float* __restrict__ eq2,
        const float* __restrict__ x3, const float* __restrict__ k3, const float* __restrict__ eq3,
        const float* __restrict__ x4, const float* __restrict__ k4,
        const float* __restrict__ ph4, const float* __restrict__ per4,
        const uint32_t* __restrict__ ordered, const uint32_t* __restrict__ startp,
        const uint32_t* __restrict__ endp,
        int N2, int N3, int N4, int C, int P, float* __restrict__ out) {
    __shared__ uint32_t sIdx[IDX_CAP];
    __shared__ float    sRed[WG];

    const int g   = blockIdx.x;
    const int tid = threadIdx.x;
    const uint32_t s0 = startp[g];
    const int cnt = (int)(endp[g] - s0);

    // Stage this graph's row-index list into LDS with the CDNA5 async path.
    const int staged = (cnt < IDX_CAP) ? cnt : IDX_CAP;
    for (int i = tid; i < staged; i += WG) {
        async_load_b32_to_lds(lds_offset_of(&sIdx[i]), ordered + s0 + i);
    }
    wait_asynccnt0();     // per-wave ASYNCcnt drain
    __syncthreads();      // make LDS visible across the workgroup

    const int lane = tid & 63;          // conformation index (C <= 64)
    const int rl   = tid >> 6;          // row-lane 0..3
    const int NRL  = WG >> 6;

    const uint32_t uN2  = (uint32_t)N2;
    const uint32_t uN23 = (uint32_t)(N2 + N3);

    float acc = 0.0f;
    if (lane < C) {
        for (int i = rl; i < cnt; i += NRL) {
            uint32_t t = (i < IDX_CAP) ? sIdx[i] : ordered[s0 + i];
            if (t < uN2) {                       // harmonic bond
                int n = (int)t;
                float k = k2[n], e = eq2[n];
                float d = x2[(size_t)n * C + lane] - e;
                acc += 0.5f * k * d * d;
            } else if (t < uN23) {               // harmonic angle
                int n = (int)(t - uN2);
                float k = k3[n], e = eq3[n];
                float d = x3[(size_t)n * C + lane] - e;
                acc += 0.5f * k * d * d;
            } else {                             // periodic torsion
                int n = (int)(t - uN23);
                float xv = x4[(size_t)n * C + lane];
                const size_t pb = (size_t)n * P;
                for (int p = 0; p < P; ++p) {
                    float kk  = k4[pb + p];      // row-uniform: broadcast loads
                    float pr  = per4[pb + p];
                    float phv = ph4[pb + p];
                    // v_cos_f32 (TRANS pipe); |arg| < ~6 rad here, accuracy ~1e-6
                    acc += kk * (1.0f + __cosf(pr * xv - phv));
                }
            }
        }
    }
    sRed[tid] = acc;
    __syncthreads();
    if (tid < 64) {
        float s = sRed[tid];
        #pragma unroll
        for (int r = 1; r < (WG >> 6); ++r) s += sRed[r * 64 + tid];
        if (tid < C) out[(size_t)g * C + tid] = s;   // single write, no atomics
    }
    (void)NRL;
}

// --------------------------- fallback: direct atomics ----------------------
// Used only if scratch is too small or C > 64. Fully correct, just slower.

__global__ void fused_atomic_kernel(
        const float* x2, const float* k2, const float* eq2, const int* gid2,
        const float* x3, const float* k3, const float* eq3, const int* gid3,
        const float* x4, const float* k4, const float* ph4, const float* per4, const int* gid4,
        int N2, int N3, int N4, int C, int P, int G, float* out) {
    const long long total = (long long)(N2 + N3 + N4) * C;
    for (long long idx = blockIdx.x * (long long)blockDim.x + threadIdx.x;
         idx < total; idx += (long long)gridDim.x * blockDim.x) {
        int t = (int)(idx / C);
        int c = (int)(idx - (long long)t * C);
        float e;
        int g;
        if (t < N2) {
            int n = t;
            float d = x2[(size_t)n * C + c] - eq2[n];
            e = 0.5f * k2[n] * d * d;
            g = gid2[n];
        } else if (t < N2 + N3) {
            int n = t - N2;
            float d = x3[(size_t)n * C + c] - eq3[n];
            e = 0.5f * k3[n] * d * d;
            g = gid3[n];
        } else {
            int n = t - N2 - N3;
            float xv = x4[(size_t)n * C + c];
            e = 0.0f;
            for (int p = 0; p < P; ++p)
                e += k4[(size_t)n * P + p] *
                     (1.0f + __cosf(per4[(size_t)n * P + p] * xv - ph4[(size_t)n * P + p]));
            g = gid4[n];
        }
        g = (g < 0) ? 0 : (g >= G ? G - 1 : g);
        atomicAdd(&out[(size_t)g * C + c], e);
    }
}

// --------------------------- launcher --------------------------------------

extern "C" void kernel_launch(void* const* d_in, const int* in_sizes, int n_in,
                              void* d_out, int out_size, void* d_ws, size_t ws_size,
                              hipStream_t stream) {
    const float* x2   = (const float*)d_in[0];
    const float* k2   = (const float*)d_in[1];
    const float* eq2  = (const float*)d_in[2];
    const int*   gid2 = (const int*)d_in[3];
    const float* x3   = (const float*)d_in[4];
    const float* k3   = (const float*)d_in[5];
    const float* eq3  = (const float*)d_in[6];
    const int*   gid3 = (const int*)d_in[7];
    const float* x4   = (const float*)d_in[8];
    const float* k4   = (const float*)d_in[9];
    const float* ph4  = (const float*)d_in[10];
    const float* per4 = (const float*)d_in[11];
    const int*   gid4 = (const int*)d_in[12];

    const int N2 = in_sizes[1];
    const int N3 = in_sizes[5];
    const int N4 = in_sizes[12];
    const int C  = in_sizes[0] / N2;       // 50
    const int P  = in_sizes[9] / N4;       // 6
    const int G  = out_size / C;           // 4096
    const int Nt = N2 + N3 + N4;
    float* out = (float*)d_out;

    const size_t need = ((size_t)3 * G + (size_t)Nt) * sizeof(uint32_t);

    if (C <= 64 && ws_size >= need) {
        uint32_t* count   = (uint32_t*)d_ws;
        uint32_t* startp  = count + G;
        uint32_t* cursor  = startp + G;
        uint32_t* ordered = cursor + G;

        zero_u32_kernel<<<(G + 255) / 256, 256, 0, stream>>>(count, G);
        hist_kernel<<<2048, 256, 0, stream>>>(gid2, gid3, gid4, N2, N3, N4, G, count);
        scan_kernel<<<1, WG, 0, stream>>>(count, startp, cursor, G);
        scatter_kernel<<<2048, 256, 0, stream>>>(gid2, gid3, gid4, N2, N3, N4, G,
                                                 cursor, ordered);
        gather_reduce_kernel<<<G, WG, 0, stream>>>(
            x2, k2, eq2, x3, k3, eq3, x4, k4, ph4, per4,
            ordered, startp, /*endp=*/cursor, N2, N3, N4, C, P, out);
    } else {
        zero_f32_kernel<<<(out_size + 255) / 256, 256, 0, stream>>>(out, out_size);
        fused_atomic_kernel<<<4096, 256, 0, stream>>>(
            x2, k2, eq2, gid2, x3, k3, eq3, gid3,
            x4, k4, ph4, per4, gid4, N2, N3, N4, C, P, G, out);
    }
}